// SelfAttention_76974403879351
// MI455X (gfx1250) — compile-verified
//
#include <hip/hip_runtime.h>
#include <hip/hip_bf16.h>

// ---------------------------------------------------------------------------
// Fused attention prefill for MI455X (gfx1250, wave32, WMMA).
// fp32 everywhere via V_WMMA_F32_16X16X4_F32: workload is HBM-bound (~15us of
// traffic at 23.3 TB/s vs ~2us of fp32 matrix math), so full-precision matrix
// ops cost nothing and match the fp32 reference exactly.
// GEMM staging uses the CDNA5 async global->LDS path (ASYNCcnt) with double
// buffering so HBM latency overlaps the WMMA stream.
// ---------------------------------------------------------------------------

typedef __attribute__((ext_vector_type(2))) float v2f;
typedef __attribute__((ext_vector_type(8))) float v8f;

#define WMMA_F32(a, b, c) \
    __builtin_amdgcn_wmma_f32_16x16x4_f32(false, (a), false, (b), (short)0, (c), false, false)

// Problem constants
#define BB   4
#define SS   1024
#define DIM  4096
#define NH   32
#define NKV  8
#define HD   128
#define MAXS 2048
#define STARTP 1024
#define KVLEN  2048

#define LDP 20   // LDS row pad (floats): 80B rows -> 16B-aligned float4 slots,
                 // stride-20 dword addressing is bank-conflict-free for 16 lanes

// CDNA5 async global->LDS copy (tracked by ASYNCcnt, not LOADcnt/DScnt).
__device__ __forceinline__ void async_ld_b128(const float* lds_ptr, const float* g) {
    unsigned lds_off = (unsigned)(unsigned long long)lds_ptr;  // low 32 bits = LDS offset
    asm volatile("global_load_async_to_lds_b128 %0, %1, off"
                 :: "v"(lds_off), "v"(g) : "memory");
}
__device__ __forceinline__ void wait_async0() {
    asm volatile("s_wait_asynccnt 0x0" ::: "memory");
}

// ---------------------------------------------------------------------------
// GEMM: C[M,N] = A[M,K] * B[N,K]^T   (A,B row-major, K-contiguous -> NT gemm)
// Block tile 128x64, BK=16, 256 threads = 8 waves in 4x2 grid, each wave 32x32.
// Double-buffered async-LDS staging.
// ---------------------------------------------------------------------------
__global__ __launch_bounds__(256)
void gemm_nt_f32(const float* __restrict__ A, const float* __restrict__ B,
                 float* __restrict__ C, int M, int N, int K)
{
    __shared__ float As[2][128 * LDP];
    __shared__ float Bs[2][64 * LDP];

    const int t    = threadIdx.x;
    const int lane = t & 31;
    const int w    = t >> 5;       // wave 0..7
    const int wm   = w >> 1;       // 0..3 (M)
    const int wn   = w & 1;        // 0..1 (N)
    const int m0   = blockIdx.y * 128;
    const int n0   = blockIdx.x * 64;
    const int fm   = lane & 15;            // fragment row/col within 16
    const int jb   = (lane >> 4) << 1;     // K sub-offset {0,2}

    // Per-thread staging slots (float4 granularity).
    // A chunk: 128 rows x 16 K = 512 float4 -> 2 per thread.
    const int rA0 = t >> 2,           cA0 = (t & 3) << 2;
    const int rA1 = (t + 256) >> 2,   cA1 = cA0;          // (t+256)&3 == t&3
    // B chunk: 64 rows x 16 K = 256 float4 -> 1 per thread.
    const int rB  = t >> 2,           cB  = (t & 3) << 2;

    v8f acc[2][2];
    #pragma unroll
    for (int i = 0; i < 2; ++i)
        #pragma unroll
        for (int j = 0; j < 2; ++j)
            acc[i][j] = (v8f)0.0f;

    const int nchunks = K >> 4;

    // preload chunk 0 into buffer 0
    {
        async_ld_b128(&As[0][rA0 * LDP + cA0], A + (size_t)(m0 + rA0) * K + cA0);
        async_ld_b128(&As[0][rA1 * LDP + cA1], A + (size_t)(m0 + rA1) * K + cA1);
        async_ld_b128(&Bs[0][rB  * LDP + cB ], B + (size_t)(n0 + rB ) * K + cB );
    }

    for (int i = 0; i < nchunks; ++i) {
        const int buf = i & 1;
        wait_async0();        // my async writes have landed in LDS
        __syncthreads();      // everyone's have

        if (i + 1 < nchunks) {
            const int kc = (i + 1) << 4;
            async_ld_b128(&As[buf ^ 1][rA0 * LDP + cA0], A + (size_t)(m0 + rA0) * K + kc + cA0);
            async_ld_b128(&As[buf ^ 1][rA1 * LDP + cA1], A + (size_t)(m0 + rA1) * K + kc + cA1);
            async_ld_b128(&Bs[buf ^ 1][rB  * LDP + cB ], B + (size_t)(n0 + rB ) * K + kc + cB );
        }

        #pragma unroll
        for (int kk = 0; kk < 4; ++kk) {
            v2f a0 = *(const v2f*)(&As[buf][(wm * 32 +      fm) * LDP + kk * 4 + jb]);
            v2f a1 = *(const v2f*)(&As[buf][(wm * 32 + 16 + fm) * LDP + kk * 4 + jb]);
            v2f b0 = *(const v2f*)(&Bs[buf][(wn * 32 +      fm) * LDP + kk * 4 + jb]);
            v2f b1 = *(const v2f*)(&Bs[buf][(wn * 32 + 16 + fm) * LDP + kk * 4 + jb]);
            acc[0][0] = WMMA_F32(a0, b0, acc[0][0]);
            acc[0][1] = WMMA_F32(a0, b1, acc[0][1]);
            acc[1][0] = WMMA_F32(a1, b0, acc[1][0]);
            acc[1][1] = WMMA_F32(a1, b1, acc[1][1]);
        }
        // No trailing barrier needed: every ds_load above is consumed by a WMMA
        // (dscnt drained) before the next iteration's barrier, and the next
        // async writes target the other buffer.
    }

    // epilogue: C/D layout -> row = v + 8*(lane>=16), col = lane&15
    const int half8 = (lane >> 4) << 3;
    #pragma unroll
    for (int mt = 0; mt < 2; ++mt)
        #pragma unroll
        for (int nt = 0; nt < 2; ++nt)
            #pragma unroll
            for (int v = 0; v < 8; ++v) {
                int row = m0 + wm * 32 + mt * 16 + v + half8;
                int col = n0 + wn * 32 + nt * 16 + fm;
                C[(size_t)row * N + col] = acc[mt][nt][v];
            }
}

// ---------------------------------------------------------------------------
// RoPE on xk AND xv (reference ropes v as well), in place on workspace copies.
// Layout (B,S,NKV,HD); one thread per even/odd pair.
// ---------------------------------------------------------------------------
__global__ __launch_bounds__(256)
void rope_kernel(float* __restrict__ kx, float* __restrict__ vx,
                 const float* __restrict__ cosb, const float* __restrict__ sinb)
{
    const int total = BB * SS * NKV * (HD / 2);   // 2,097,152 pairs
    int idx = blockIdx.x * blockDim.x + threadIdx.x;
    if (idx >= total) return;
    int i = idx & 63;               // freq index 0..63
    int s = (idx >> 9) & 1023;      // position within sequence
    float c  = cosb[s * 64 + i];
    float sn = sinb[s * 64 + i];

    float2 k2 = *(const float2*)(kx + 2 * (size_t)idx);
    float2 ko; ko.x = k2.x * c - k2.y * sn; ko.y = k2.x * sn + k2.y * c;
    *(float2*)(kx + 2 * (size_t)idx) = ko;

    float2 vv = *(const float2*)(vx + 2 * (size_t)idx);
    float2 vo; vo.x = vv.x * c - vv.y * sn; vo.y = vv.x * sn + vv.y * c;
    *(float2*)(vx + 2 * (size_t)idx) = vo;
}

// ---------------------------------------------------------------------------
// Flash attention: one wave per (16-query tile, head). kv_len = 2048, no mask
// (reference has none): pos < 1024 reads the input caches, pos >= 1024 reads
// the freshly-roped workspace K/V (inputs are never mutated).
// ---------------------------------------------------------------------------
__global__ __launch_bounds__(32)
void attn_kernel(const float* __restrict__ q,      // (B,S,NH,HD)
                 const float* __restrict__ knew,   // (B,S,NKV,HD) roped
                 const float* __restrict__ vnew,   // (B,S,NKV,HD) roped
                 const float* __restrict__ kcache, // (B,MAXS,NKV,HD)
                 const float* __restrict__ vcache,
                 float* __restrict__ out)          // (B,S,NH,HD)
{
    const int lane = threadIdx.x & 31;
    const int qt   = blockIdx.x;      // 0..63 query tile
    const int h    = blockIdx.y;      // 0..31 head
    const int b    = blockIdx.z;
    const int g    = h >> 2;          // kv group (rep = 4)
    const int fm   = lane & 15;
    const int jb   = (lane >> 4) << 1;
    const int half8 = (lane >> 4) << 3;

    __shared__ float pshare[16 * 18];

    // Q tile as 32 K-step A-fragments held in VGPRs (16 rows x 128 dims)
    v2f qf[32];
    const float* qrow = q + (((size_t)(b * SS + qt * 16 + fm)) * NH + h) * HD;
    #pragma unroll
    for (int kk = 0; kk < 32; ++kk)
        qf[kk] = *(const v2f*)(qrow + kk * 4 + jb);

    v8f o[8];
    float rm[8], rl[8];
    #pragma unroll
    for (int nt = 0; nt < 8; ++nt) o[nt] = (v8f)0.0f;
    #pragma unroll
    for (int v = 0; v < 8; ++v) { rm[v] = -1.0e30f; rl[v] = 0.0f; }

    const float scale = 0.08838834764831845f;   // 1/sqrt(128)

    for (int t = 0; t < KVLEN / 16; ++t) {
        // ---- S = Q * K^T for this 16-key tile (this lane's key = fm)
        int kpos = t * 16 + fm;
        const float* kp = (kpos < STARTP)
            ? kcache + (((size_t)(b * MAXS + kpos)) * NKV + g) * HD
            : knew   + (((size_t)(b * SS + (kpos - STARTP))) * NKV + g) * HD;

        v8f s = (v8f)0.0f;
        #pragma unroll
        for (int kk = 0; kk < 32; ++kk) {
            v2f bf = *(const v2f*)(kp + kk * 4 + jb);
            s = WMMA_F32(qf[kk], bf, s);
        }

        // ---- online softmax (row reductions across 16 lanes of each half)
        float p[8];
        #pragma unroll
        for (int v = 0; v < 8; ++v) {
            float sv = s[v] * scale;
            float mx = sv;
            mx = fmaxf(mx, __shfl_xor(mx, 1, 32));
            mx = fmaxf(mx, __shfl_xor(mx, 2, 32));
            mx = fmaxf(mx, __shfl_xor(mx, 4, 32));
            mx = fmaxf(mx, __shfl_xor(mx, 8, 32));
            float mnew  = fmaxf(rm[v], mx);
            float alpha = __expf(rm[v] - mnew);
            float pv    = __expf(sv - mnew);
            float sum = pv;
            sum += __shfl_xor(sum, 1, 32);
            sum += __shfl_xor(sum, 2, 32);
            sum += __shfl_xor(sum, 4, 32);
            sum += __shfl_xor(sum, 8, 32);
            rl[v] = rl[v] * alpha + sum;
            rm[v] = mnew;
            p[v]  = pv;
            #pragma unroll
            for (int nt = 0; nt < 8; ++nt) o[nt][v] *= alpha;   // rescale row v
        }

        // ---- P: C-layout -> A-layout via padded LDS tile (single wave: LDS in-order)
        #pragma unroll
        for (int v = 0; v < 8; ++v)
            pshare[(v + half8) * 18 + fm] = p[v];

        v2f pf[4];
        #pragma unroll
        for (int kk = 0; kk < 4; ++kk)
            pf[kk] = *(const v2f*)(&pshare[fm * 18 + kk * 4 + jb]);

        // ---- O += P * V
        #pragma unroll
        for (int nt = 0; nt < 8; ++nt) {
            int dim = nt * 16 + fm;
            #pragma unroll
            for (int kk = 0; kk < 4; ++kk) {
                int kl = kk * 4 + jb;           // key-in-tile for .x; .y is kl+1
                int p0 = t * 16 + kl;
                const float* vp0 = (p0 < STARTP)
                    ? vcache + (((size_t)(b * MAXS + p0)) * NKV + g) * HD
                    : vnew   + (((size_t)(b * SS + (p0 - STARTP))) * NKV + g) * HD;
                const float* vp1 = (p0 + 1 < STARTP)
                    ? vcache + (((size_t)(b * MAXS + p0 + 1)) * NKV + g) * HD
                    : vnew   + (((size_t)(b * SS + (p0 + 1 - STARTP))) * NKV + g) * HD;
                v2f vf;
                vf.x = vp0[dim];
                vf.y = vp1[dim];
                o[nt] = WMMA_F32(pf[kk], vf, o[nt]);
            }
        }
    }

    // ---- normalize and store (B,S,NH,HD)
    #pragma unroll
    for (int v = 0; v < 8; ++v) {
        float inv = 1.0f / rl[v];
        int row = qt * 16 + v + half8;
        float* orow = out + (((size_t)(b * SS + row)) * NH + h) * HD;
        #pragma unroll
        for (int nt = 0; nt < 8; ++nt)
            orow[nt * 16 + fm] = o[nt][v] * inv;
    }
}

// ---------------------------------------------------------------------------
// Host launcher. d_in order: x, wq, wk, wv, wo, k_cache, v_cache,
//                            theta_cos, theta_sin, start_pos
// ---------------------------------------------------------------------------
extern "C" void kernel_launch(void* const* d_in, const int* in_sizes, int n_in,
                              void* d_out, int out_size, void* d_ws, size_t ws_size,
                              hipStream_t stream) {
    (void)in_sizes; (void)n_in; (void)out_size; (void)ws_size;

    const float* x      = (const float*)d_in[0];
    const float* wq     = (const float*)d_in[1];
    const float* wk     = (const float*)d_in[2];
    const float* wv     = (const float*)d_in[3];
    const float* wo     = (const float*)d_in[4];
    const float* kcache = (const float*)d_in[5];
    const float* vcache = (const float*)d_in[6];
    const float* tcos   = (const float*)d_in[7];
    const float* tsin   = (const float*)d_in[8];

    const size_t M = (size_t)BB * SS;          // 4096 rows
    float* xq   = (float*)d_ws;                            // 4096*4096
    float* xk   = xq + M * DIM;                            // 4096*1024
    float* xv   = xk + M * (size_t)(NKV * HD);             // 4096*1024
    float* attn = xv + M * (size_t)(NKV * HD);             // 4096*4096
    // total ws: ~168 MB of fp32

    // 1-3) QKV projections (C = X * W^T)
    gemm_nt_f32<<<dim3(DIM / 64, M / 128), 256, 0, stream>>>(x, wq, xq, (int)M, DIM, DIM);
    gemm_nt_f32<<<dim3((NKV * HD) / 64, M / 128), 256, 0, stream>>>(x, wk, xk, (int)M, NKV * HD, DIM);
    gemm_nt_f32<<<dim3((NKV * HD) / 64, M / 128), 256, 0, stream>>>(x, wv, xv, (int)M, NKV * HD, DIM);

    // 4) RoPE on k and v
    {
        int pairs = BB * SS * NKV * (HD / 2);
        rope_kernel<<<(pairs + 255) / 256, 256, 0, stream>>>(xk, xv, tcos, tsin);
    }

    // 5) attention
    attn_kernel<<<dim3(SS / 16, NH, BB), 32, 0, stream>>>(xq, xk, xv, kcache, vcache, attn);

    // 6) output projection into d_out
    gemm_nt_f32<<<dim3(DIM / 64, M / 128), 256, 0, stream>>>(attn, wo, (float*)d_out, (int)M, DIM, DIM);
}